// InstHead_5291399708799
// MI455X (gfx1250) — compile-verified
//
#include <hip/hip_runtime.h>

typedef float v2f __attribute__((ext_vector_type(2)));
typedef float v8f __attribute__((ext_vector_type(8)));

#define KCLS 20
#define NBASE 12
#define PROPS 1024
#define PROPW 22   // 20 sem sums + binary sum + count

// ws layout (floats):
//  [0,32)     sum f        [32,1056) sum f f^T (32x32)
//  [1056,1088) BN scale    [1088,1120) BN shift
//  [1120,1120+PROPS*PROPW) proposal accumulators
#define WS_SF    0
#define WS_SFF   32
#define WS_SCL   1056
#define WS_SHF   1088
#define WS_PROP  1120
#define WS_TOTAL (WS_PROP + PROPS * PROPW)

__device__ __forceinline__ v8f wmma4(v2f a, v2f b, v8f c) {
  // D = A(16x4, f32) * B(4x16, f32) + C(16x16, f32)
  return __builtin_amdgcn_wmma_f32_16x16x4_f32(false, a, false, b, (short)0, c,
                                               false, false);
}

// Intra-wave LDS producer->consumer ordering: hardware keeps per-wave LDS ops
// in order; this only pins compiler scheduling.
__device__ __forceinline__ void wave_sync() {
  __builtin_amdgcn_wave_barrier();
  asm volatile("" ::: "memory");
}

// CDNA5 async copy global -> LDS (ASYNCcnt-tracked, no VGPR round trip).
__device__ __forceinline__ void async_ld_b128(unsigned lds_off, const float* g) {
  asm volatile("global_load_async_to_lds_b128 %0, %1, off"
               :: "v"(lds_off), "v"(g)
               : "memory");
}

__global__ __launch_bounds__(256) void zero_ws(float* ws, int n) {
  int i = blockIdx.x * 256 + threadIdx.x;
  if (i < n) ws[i] = 0.0f;
}

// ---- Pass A: sum(f) and F^T F via WMMA; branchless (tail done in finalize) ----
__global__ __launch_bounds__(256) void stats_kernel(
    const float* __restrict__ feats, float* __restrict__ ws, int N) {
  int tid = threadIdx.x;
  int wave = tid >> 5, lane = tid & 31;
  int hi = lane >> 4, lm = lane & 15;
  v8f a00 = {}, a01 = {}, a10 = {}, a11 = {};
  float sf0 = 0.f, sf1 = 0.f;
  long gw = (long)blockIdx.x * 8 + wave;
  long stride = (long)gridDim.x * 8;
  long nFull = (long)N >> 2;  // full groups of 4 points
  for (long it = gw; it < nFull; it += stride) {
    const float* rp = feats + (it * 4 + 2 * hi) * 32 + lm;
    v2f f0, f1;
    f0.x = rp[0];  f0.y = rp[32];   // rows p+2hi, p+2hi+1 : col lm
    f1.x = rp[16]; f1.y = rp[48];   // same rows           : col lm+16
    sf0 += f0.x + f0.y;
    sf1 += f1.x + f1.y;
    a00 = wmma4(f0, f0, a00);   // Sff[0:16,0:16]
    a01 = wmma4(f0, f1, a01);   // Sff[0:16,16:32]
    a10 = wmma4(f1, f0, a10);   // Sff[16:32,0:16]
    a11 = wmma4(f1, f1, a11);   // Sff[16:32,16:32]
  }
  atomicAdd(&ws[WS_SF + lm], sf0);
  atomicAdd(&ws[WS_SF + 16 + lm], sf1);
#pragma unroll
  for (int r = 0; r < 8; ++r) {
    int m = r + 8 * hi;
    atomicAdd(&ws[WS_SFF + m * 32 + lm], a00[r]);
    atomicAdd(&ws[WS_SFF + m * 32 + 16 + lm], a01[r]);
    atomicAdd(&ws[WS_SFF + (16 + m) * 32 + lm], a10[r]);
    atomicAdd(&ws[WS_SFF + (16 + m) * 32 + 16 + lm], a11[r]);
  }
}

// ---- Pass B: add tail (<4 pts), fold BN (eps=1e-4) into per-channel affine ----
__global__ void finalize_bn(const float* __restrict__ feats,
                            const float* __restrict__ W1,
                            const float* __restrict__ b1,
                            const float* __restrict__ g1,
                            const float* __restrict__ be1,
                            float* __restrict__ ws, int N) {
  __shared__ float sff[1024];
  __shared__ float sf[32];
  int j = threadIdx.x;  // launched with 32 threads
  sf[j] = ws[WS_SF + j];
  for (int i = j; i < 1024; i += 32) sff[i] = ws[WS_SFF + i];
  __syncthreads();
  int tail = (N >> 2) << 2;
  for (int p = tail; p < N; ++p) {  // thread j owns column j
    float fj = feats[(long)p * 32 + j];
    sf[j] += fj;
    for (int a = 0; a < 32; ++a) sff[a * 32 + j] += feats[(long)p * 32 + a] * fj;
  }
  __syncthreads();
  float invN = 1.0f / (float)N;
  float sdot = 0.f, q = 0.f;
  for (int a = 0; a < 32; ++a) {
    float wa = W1[a * 32 + j];
    sdot += sf[a] * wa;
    float row = 0.f;
    for (int b = 0; b < 32; ++b) row += sff[a * 32 + b] * W1[b * 32 + j];
    q += wa * row;
  }
  float bj = b1[j];
  float mean = sdot * invN + bj;
  float Et2 = (q + 2.f * bj * sdot) * invN + bj * bj;
  float var = Et2 - mean * mean;
  float s = g1[j] * rsqrtf(var + 1e-4f);
  ws[WS_SCL + j] = s;
  ws[WS_SHF + j] = be1[j] - mean * s;
}

// ---- Pass C: fused per-point heads; per-wave tiles, async double-buffer ----
__global__ __launch_bounds__(256) void fused_point_kernel(
    const float* __restrict__ feats, const float* __restrict__ sem,
    const float* __restrict__ binp, const float* __restrict__ W1,
    const float* __restrict__ b1, const float* __restrict__ W2,
    const float* __restrict__ b2, const float* __restrict__ Wm1,
    const float* __restrict__ bm1, const float* __restrict__ Wm2,
    const float* __restrict__ bm2, const float* __restrict__ Wi,
    const float* __restrict__ bi, const float* __restrict__ ws,
    float* __restrict__ outScores, float* __restrict__ outOff,
    float* __restrict__ outMask, float* __restrict__ outIou, int N) {
  __shared__ __align__(16) float sW1[1024], sWm1[1024];
  __shared__ __align__(16) float sW2[96];
  __shared__ float sWm2[32], sWi[32], sB1[32], sBm1[32], sS[32], sC[32];
  __shared__ __align__(16) float sF[8][2][512];  // double-buffered feats tiles
  __shared__ __align__(16) float sH[8][512];     // hidden tiles

  int tid = threadIdx.x;
  for (int i = tid; i < 1024; i += 256) { sW1[i] = W1[i]; sWm1[i] = Wm1[i]; }
  if (tid < 96) sW2[tid] = W2[tid];
  if (tid < 32) {
    sWm2[tid] = Wm2[tid]; sWi[tid] = Wi[tid];
    sB1[tid] = b1[tid];   sBm1[tid] = bm1[tid];
    sS[tid] = ws[WS_SCL + tid]; sC[tid] = ws[WS_SHF + tid];
  }
  __syncthreads();  // weights visible to all waves; only block-wide barrier

  float b2v0 = b2[0], b2v1 = b2[1], b2v2 = b2[2];
  float bm2v = bm2[0], biv = bi[0];

  int wave = tid >> 5, lane = tid & 31;
  int hi = lane >> 4, lm = lane & 15;
  float* buf0 = sF[wave][0];
  float* buf1 = sF[wave][1];
  float* hT = sH[wave];

  long numTiles = ((long)N + 15) >> 4;
  long gw = (long)blockIdx.x * 8 + wave;
  long stride = (long)gridDim.x * 8;
  long maxv4 = ((long)N * 32) >> 2;

  // prologue: issue first tile into buf0 (clamped addresses -> always 4 ops)
  if (gw < numTiles) {
    long b4 = gw << 7;  // tile*128 float4s
#pragma unroll
    for (int it2 = 0; it2 < 4; ++it2) {
      int sl = lane + 32 * it2;
      long idx = b4 + sl;
      if (idx >= maxv4) idx = 0;
      async_ld_b128((unsigned)(uintptr_t)(buf0 + 4 * sl), feats + idx * 4);
    }
  }

  int cur = 0;
  for (long t = gw; t < numTiles; t += stride) {
    long tn = t + stride;
    bool pre = (tn < numTiles);
    if (pre) {  // prefetch next tile into the other buffer
      float* fN = cur ? buf0 : buf1;
      long b4 = tn << 7;
#pragma unroll
      for (int it2 = 0; it2 < 4; ++it2) {
        int sl = lane + 32 * it2;
        long idx = b4 + sl;
        if (idx >= maxv4) idx = 0;
        async_ld_b128((unsigned)(uintptr_t)(fN + 4 * sl), feats + idx * 4);
      }
      // async loads complete in order: <=4 outstanding => current tile landed
      asm volatile("s_wait_asynccnt 0x4" ::: "memory");
    } else {
      asm volatile("s_wait_asynccnt 0x0" ::: "memory");
    }
    wave_sync();

    float* fT = cur ? buf1 : buf0;
    long base = t << 4;
    long p = base + lm;  // point handled by this lane's half-pair

    // ---- iou: split 32 cols across half-waves, combine with shfl_xor ----
    {
      float io = 0.f;
#pragma unroll
      for (int c = 0; c < 16; ++c)
        io += fT[lm * 32 + 16 * hi + c] * sWi[16 * hi + c];
      io += __shfl_xor(io, 16);
      if (lane < 16 && p < N) outIou[p] = io + biv;
    }

    // ---- open-vocab score renorm (lanes 0..15, one point each) ----
    if (lane < 16 && p < N) {
      const float* srow = sem + p * (long)KCLS;
      float sr[KCLS];
#pragma unroll
      for (int k = 0; k < KCLS; ++k) sr[k] = srow[k];
      float m1 = sr[0], m2 = sr[NBASE];
#pragma unroll
      for (int k = 1; k < NBASE; ++k) m1 = fmaxf(m1, sr[k]);
#pragma unroll
      for (int k = NBASE + 1; k < KCLS; ++k) m2 = fmaxf(m2, sr[k]);
      float e1 = 0.f, e2 = 0.f;
#pragma unroll
      for (int k = 0; k < NBASE; ++k) { sr[k] = __expf(sr[k] - m1); e1 += sr[k]; }
#pragma unroll
      for (int k = NBASE; k < KCLS; ++k) { sr[k] = __expf(sr[k] - m2); e2 += sr[k]; }
      float sg = 1.f / (1.f + __expf(-binp[p]));
      float tot = 0.f;
#pragma unroll
      for (int k = 0; k < NBASE; ++k) { sr[k] = sr[k] / e1 * sg; tot += sr[k]; }
#pragma unroll
      for (int k = NBASE; k < KCLS; ++k) { sr[k] = sr[k] / e2 * (1.f - sg); tot += sr[k]; }
      float invTot = 1.f / tot;
      float* orow = outScores + p * (long)KCLS;
#pragma unroll
      for (int k = 0; k < KCLS; ++k) orow[k] = sr[k] * invTot;
    }

    // ---- A fragments (shared by both GEMMs) ----
    v2f a[8];
#pragma unroll
    for (int ks = 0; ks < 8; ++ks)
      a[ks] = *(const v2f*)(fT + lm * 32 + 4 * ks + 2 * hi);

    // ---- GEMM1: t = F*W1 + b1, BN-affine + ReLU -> hT ----
    {
      v8f acc0, acc1;
      float i0 = sB1[lm], i1 = sB1[16 + lm];
#pragma unroll
      for (int r = 0; r < 8; ++r) { acc0[r] = i0; acc1[r] = i1; }
#pragma unroll
      for (int ks = 0; ks < 8; ++ks) {
        int krow = 4 * ks + 2 * hi;
        v2f bA, bB;
        bA.x = sW1[krow * 32 + lm];      bA.y = sW1[(krow + 1) * 32 + lm];
        bB.x = sW1[krow * 32 + 16 + lm]; bB.y = sW1[(krow + 1) * 32 + 16 + lm];
        acc0 = wmma4(a[ks], bA, acc0);
        acc1 = wmma4(a[ks], bB, acc1);
      }
      float s0 = sS[lm], s1 = sS[16 + lm], c0 = sC[lm], c1 = sC[16 + lm];
#pragma unroll
      for (int r = 0; r < 8; ++r) {
        float v0 = acc0[r] * s0 + c0;
        float v1 = acc1[r] * s1 + c1;
        int row = r + 8 * hi;
        hT[row * 32 + lm]      = v0 > 0.f ? v0 : 0.f;
        hT[row * 32 + 16 + lm] = v1 > 0.f ? v1 : 0.f;
      }
    }
    wave_sync();

    // ---- pt_offsets = h @ W2 + b2 (split cols, shfl-combine) ----
    {
      float o0 = 0.f, o1 = 0.f, o2 = 0.f;
#pragma unroll
      for (int c = 0; c < 16; ++c) {
        int col = 16 * hi + c;
        float h = hT[lm * 32 + col];
        o0 += h * sW2[col * 3 + 0];
        o1 += h * sW2[col * 3 + 1];
        o2 += h * sW2[col * 3 + 2];
      }
      o0 += __shfl_xor(o0, 16);
      o1 += __shfl_xor(o1, 16);
      o2 += __shfl_xor(o2, 16);
      if (lane < 16 && p < N) {
        outOff[p * 3 + 0] = o0 + b2v0;
        outOff[p * 3 + 1] = o1 + b2v1;
        outOff[p * 3 + 2] = o2 + b2v2;
      }
    }
    wave_sync();

    // ---- GEMM2: m1 = relu(F*Wm1 + bm1) -> hT ----
    {
      v8f acc0, acc1;
      float i0 = sBm1[lm], i1 = sBm1[16 + lm];
#pragma unroll
      for (int r = 0; r < 8; ++r) { acc0[r] = i0; acc1[r] = i1; }
#pragma unroll
      for (int ks = 0; ks < 8; ++ks) {
        int krow = 4 * ks + 2 * hi;
        v2f bA, bB;
        bA.x = sWm1[krow * 32 + lm];      bA.y = sWm1[(krow + 1) * 32 + lm];
        bB.x = sWm1[krow * 32 + 16 + lm]; bB.y = sWm1[(krow + 1) * 32 + 16 + lm];
        acc0 = wmma4(a[ks], bA, acc0);
        acc1 = wmma4(a[ks], bB, acc1);
      }
#pragma unroll
      for (int r = 0; r < 8; ++r) {
        int row = r + 8 * hi;
        hT[row * 32 + lm]      = acc0[r] > 0.f ? acc0[r] : 0.f;
        hT[row * 32 + 16 + lm] = acc1[r] > 0.f ? acc1[r] : 0.f;
      }
    }
    wave_sync();

    // ---- mask = m1 @ Wm2 + bm2 (split cols, shfl-combine) ----
    {
      float mk = 0.f;
#pragma unroll
      for (int c = 0; c < 16; ++c) {
        int col = 16 * hi + c;
        mk += hT[lm * 32 + col] * sWm2[col];
      }
      mk += __shfl_xor(mk, 16);
      if (lane < 16 && p < N) outMask[p] = mk + bm2v;
    }
    cur ^= 1;
  }
}

// ---- Pass D: proposal scatter-mean (LDS-staged, two 512-proposal passes) ----
__global__ __launch_bounds__(256) void prop_accum(
    const float* __restrict__ sem, const float* __restrict__ binp,
    const int* __restrict__ idx, const int* __restrict__ pid,
    float* __restrict__ ws, long M) {
  __shared__ float acc[512 * PROPW];
  for (int pass = 0; pass < 2; ++pass) {
    for (int i = threadIdx.x; i < 512 * PROPW; i += 256) acc[i] = 0.f;
    __syncthreads();
    int lo = pass * 512, hiP = lo + 512;
    long start = (long)blockIdx.x * 256 + threadIdx.x;
    long stride = (long)gridDim.x * 256;
    for (long i = start; i < M; i += stride) {
      int pr = pid[i];
      if (pr < lo || pr >= hiP) continue;
      int p = idx[i];
      float* a = acc + (pr - lo) * PROPW;
      const float* srow = sem + (long)p * KCLS;
#pragma unroll
      for (int k = 0; k < KCLS; ++k) atomicAdd(&a[k], srow[k]);
      atomicAdd(&a[KCLS], binp[p]);
      atomicAdd(&a[KCLS + 1], 1.f);
    }
    __syncthreads();
    for (int i = threadIdx.x; i < 512 * PROPW; i += 256) {
      float v = acc[i];
      if (v != 0.f) atomicAdd(&ws[WS_PROP + lo * PROPW + i], v);
    }
    __syncthreads();
  }
}

__global__ void prop_final(const float* __restrict__ ws,
                           float* __restrict__ outSem,
                           float* __restrict__ outBin) {
  int p = blockIdx.x * blockDim.x + threadIdx.x;
  if (p >= PROPS) return;
  const float* a = ws + WS_PROP + p * PROPW;
  float c = a[KCLS + 1];
  c = c < 1.f ? 1.f : c;
  float inv = 1.f / c;
#pragma unroll
  for (int k = 2; k < KCLS; ++k) outSem[p * (KCLS - 2) + (k - 2)] = a[k] * inv;
  outBin[p] = a[KCLS] * inv;
}

extern "C" void kernel_launch(void* const* d_in, const int* in_sizes, int n_in,
                              void* d_out, int out_size, void* d_ws, size_t ws_size,
                              hipStream_t stream) {
  const float* feats = (const float*)d_in[0];
  const float* sem   = (const float*)d_in[1];
  const float* binp  = (const float*)d_in[2];
  const int* pidx    = (const int*)d_in[3];
  const int* pid     = (const int*)d_in[4];
  const float* W1  = (const float*)d_in[5];
  const float* b1  = (const float*)d_in[6];
  const float* g1  = (const float*)d_in[7];
  const float* be1 = (const float*)d_in[8];
  const float* W2  = (const float*)d_in[9];
  const float* b2  = (const float*)d_in[10];
  const float* Wm1 = (const float*)d_in[11];
  const float* bm1 = (const float*)d_in[12];
  const float* Wm2 = (const float*)d_in[13];
  const float* bm2 = (const float*)d_in[14];
  const float* Wi  = (const float*)d_in[15];
  const float* bi  = (const float*)d_in[16];

  int N = in_sizes[0] / 32;
  long M = in_sizes[3];
  float* ws = (float*)d_ws;
  float* out = (float*)d_out;

  float* outScores = out;                                  // [N,20]
  float* outOff    = outScores + (long)N * KCLS;           // [N,3]
  float* outPSem   = outOff + (long)N * 3;                 // [P,18]
  float* outPBin   = outPSem + (long)PROPS * (KCLS - 2);   // [P,1]
  float* outMask   = outPBin + PROPS;                      // [N,1]
  float* outIou    = outMask + N;                          // [N,1]

  zero_ws<<<(WS_TOTAL + 255) / 256, 256, 0, stream>>>(ws, WS_TOTAL);

  stats_kernel<<<1024, 256, 0, stream>>>(feats, ws, N);
  finalize_bn<<<1, 32, 0, stream>>>(feats, W1, b1, g1, be1, ws, N);

  long numTiles = ((long)N + 15) >> 4;
  int fblocks = (int)((numTiles + 7) / 8);
  if (fblocks > 1024) fblocks = 1024;
  if (fblocks < 1) fblocks = 1;
  fused_point_kernel<<<fblocks, 256, 0, stream>>>(
      feats, sem, binp, W1, b1, W2, b2, Wm1, bm1, Wm2, bm2, Wi, bi, ws,
      outScores, outOff, outMask, outIou, N);

  prop_accum<<<128, 256, 0, stream>>>(sem, binp, pidx, pid, ws, M);
  prop_final<<<(PROPS + 255) / 256, 256, 0, stream>>>(ws, outPSem, outPBin);
}